// EMGNN_73967926771800
// MI455X (gfx1250) — compile-verified
//
#include <hip/hip_runtime.h>
#include <math.h>

// ---------------------------------------------------------------- constants
constexpr int kB    = 16;
constexpr int kN    = 200;
constexpr int kTin  = 70;
constexpr int kRes  = 32;   // RES_CH
constexpr int kConv = 32;   // CONV_CH
constexpr int kSkip = 64;   // SKIP_CH
constexpr int kD    = 32;   // D_DYN
constexpr int kStat = 40;   // D_STAT
constexpr int kTlen = 64;   // T_IN - (7-1)
constexpr int kIntv = 16;
constexpr int kNCh  = 4;    // T_LEN / INTERVAL
constexpr float kAlpha = 0.05f;
constexpr float kEps   = 1e-5f;

constexpr int kXnElems  = kB * kRes * kN * kTlen;   // 6,553,600
constexpr int kPerB     = kRes * kN * kTlen;        // 409,600

// workspace layout (floats)
constexpr size_t OFF_XT    = 0;
constexpr size_t OFF_INP   = OFF_XT  + (size_t)kXnElems;          // [b][chunk][n][c]
constexpr size_t OFF_STATE = OFF_INP + (size_t)kB*kNCh*kN*kD;
constexpr size_t OFF_H     = OFF_STATE + (size_t)kB*kN*kD;
constexpr size_t OFF_AS    = OFF_H   + (size_t)kB*kN*kD;
constexpr size_t OFF_CS    = OFF_AS  + (size_t)kB*kN*kD;
constexpr size_t OFF_AM    = OFF_CS  + (size_t)kB*kN*kD;
constexpr size_t OFF_CM    = OFF_AM  + (size_t)kB*kN*kD;
constexpr size_t OFF_ADJ   = OFF_CM  + (size_t)kB*kN*kD;          // [b][v][w]
constexpr size_t OFF_HO    = OFF_ADJ + (size_t)kB*kN*kN;          // [b][96][n][16]
constexpr size_t OFF_PS    = OFF_HO  + (size_t)kB*96*kN*kIntv;
constexpr size_t OFF_PQ    = OFF_PS  + (size_t)kB*64;
constexpr size_t OFF_MV    = OFF_PQ  + (size_t)kB*64;             // mean,inv per b

// ---------------------------------------------------------------- WMMA f32
typedef float v2f __attribute__((ext_vector_type(2)));
typedef float v8f __attribute__((ext_vector_type(8)));

__device__ inline v8f wmma_f32(v2f a, v2f b, v8f c) {
  // V_WMMA_F32_16X16X4_F32: D = A(16x4) * B(4x16) + C
  return __builtin_amdgcn_wmma_f32_16x16x4_f32(
      /*neg_a=*/false, a, /*neg_b=*/false, b,
      /*c_mod=*/(short)0, c, /*reuse_a=*/false, /*reuse_b=*/false);
}

__device__ inline float sigmoidf_(float x) { return 1.0f / (1.0f + expf(-x)); }

// ---------------------------------------------------------------- kernel 1: gated dilated conv
__global__ void conv_gated_kernel(
    const float* __restrict__ x,
    const float* fw0, const float* fw1, const float* fw2, const float* fw3,
    const float* fb0, const float* fb1, const float* fb2, const float* fb3,
    const float* gw0, const float* gw1, const float* gw2, const float* gw3,
    const float* gb0, const float* gb1, const float* gb2, const float* gb3,
    float* __restrict__ xt)
{
  int gid = blockIdx.x * blockDim.x + threadIdx.x;
  if (gid >= kB * kConv * kN * kTlen) return;
  int t = gid % kTlen;
  int n = (gid / kTlen) % kN;
  int c = (gid / (kTlen * kN)) % kConv;
  int b = gid / (kTlen * kN * kConv);
  int grp = c >> 3, cc = c & 7;
  const int ktab[4] = {2, 3, 6, 7};
  int k = ktab[grp];
  const float* fw = (grp == 0) ? fw0 : (grp == 1) ? fw1 : (grp == 2) ? fw2 : fw3;
  const float* gw = (grp == 0) ? gw0 : (grp == 1) ? gw1 : (grp == 2) ? gw2 : gw3;
  float accF = ((grp == 0) ? fb0 : (grp == 1) ? fb1 : (grp == 2) ? fb2 : fb3)[cc];
  float accG = ((grp == 0) ? gb0 : (grp == 1) ? gb1 : (grp == 2) ? gb2 : gb3)[cc];
  int t0 = t + 7 - k;  // all taps end at input time t+6
  for (int ci = 0; ci < kRes; ++ci) {
    const float* xp  = x + (((size_t)b * kRes + ci) * kN + n) * kTin + t0;
    const float* fwp = fw + (cc * kRes + ci) * k;
    const float* gwp = gw + (cc * kRes + ci) * k;
    for (int j = 0; j < k; ++j) {
      float xv = xp[j];
      accF = fmaf(xv, fwp[j], accF);
      accG = fmaf(xv, gwp[j], accG);
    }
  }
  xt[gid] = tanhf(accF) * sigmoidf_(accG);
}

// ---------------------------------------------------------------- kernel 2: skip conv (WMMA f32 GEMM)
// out[b,o,n] = sum_{c,t} xt[b,c,n,t] * skip_w[o,c,t] + skip_b[o]
// rows m=(b*200+n) in [0,3200), cols=o in [0,64), K=(c*64+t) in [0,2048)
__global__ void skip_wmma_kernel(const float* __restrict__ xt,
                                 const float* __restrict__ skip_w,
                                 const float* __restrict__ skip_b,
                                 float* __restrict__ out)
{
  int lane = threadIdx.x & 31, l16 = lane & 15, half = lane >> 4;
  int rt = blockIdx.x;  // 0..199
  int ct = blockIdx.y;  // 0..3
  int rowA = rt * 16 + l16;
  int bA = rowA / kN, nA = rowA % kN;
  int col = ct * 16 + l16;  // o for B / D
  const float* arow = xt + ((size_t)bA * kConv) * kN * kTlen + (size_t)nA * kTlen;
  v8f acc = {};
  for (int k0 = 0; k0 < kConv * kTlen; k0 += 4) {
    int ka = k0 + 2 * half;
    v2f a, bv;
    a.x = arow[(ka >> 6) * kN * kTlen + (ka & 63)];
    a.y = arow[((ka + 1) >> 6) * kN * kTlen + ((ka + 1) & 63)];
    bv.x = skip_w[col * 2048 + ka];
    bv.y = skip_w[col * 2048 + ka + 1];
    acc = wmma_f32(a, bv, acc);
  }
  for (int r = 0; r < 8; ++r) {
    int m = rt * 16 + r + 8 * half;
    int bo = m / kN, no = m % kN;
    out[((size_t)bo * kSkip + col) * kN + no] = acc[r] + skip_b[col];
  }
}

// ---------------------------------------------------------------- kernel 3: per-chunk time means
__global__ void chunk_means_kernel(const float* __restrict__ xt, float* __restrict__ inp)
{
  int gid = blockIdx.x * blockDim.x + threadIdx.x;
  if (gid >= kB * kNCh * kN * kD) return;
  int c = gid % kD;
  int n = (gid / kD) % kN;
  int ch = (gid / (kD * kN)) % kNCh;
  int b = gid / (kD * kN * kNCh);
  const float* p = xt + (((size_t)b * kConv + c) * kN + n) * kTlen + ch * kIntv;
  float s = 0.f;
  for (int t = 0; t < kIntv; ++t) s += p[t];
  inp[gid] = s * (1.0f / kIntv);
}

// ---------------------------------------------------------------- kernel 4: initial state from static features
__global__ void state_init_kernel(const float* __restrict__ st, const float* __restrict__ w,
                                  const float* __restrict__ bias, float* __restrict__ state)
{
  int gid = blockIdx.x * blockDim.x + threadIdx.x;
  if (gid >= kB * kN * kD) return;
  int c = gid % kD;
  int n = (gid / kD) % kN;
  float s = bias[c];
  for (int e = 0; e < kStat; ++e) s = fmaf(st[n * kStat + e], w[c * kStat + e], s);
  state[gid] = s;
}

// ---------------------------------------------------------------- kernel 5: GRU step (per chunk)
__global__ void gru_kernel(const float* __restrict__ inp, float* __restrict__ state,
                           const float* __restrict__ rz_w, const float* __restrict__ rz_b,
                           const float* __restrict__ hc_w, const float* __restrict__ hc_b,
                           float* __restrict__ h, int chunk)
{
  int gid = blockIdx.x * blockDim.x + threadIdx.x;
  if (gid >= kB * kN) return;
  int n = gid % kN, b = gid / kN;
  const float* ip = inp + (((size_t)b * kNCh + chunk) * kN + n) * kD;
  float* sp = state + (size_t)gid * kD;
  float xin[kD], st[kD], r[kD];
  for (int e = 0; e < kD; ++e) { xin[e] = ip[e]; st[e] = sp[e]; }
  for (int o = 0; o < kD; ++o) {
    float acc = rz_b[o];
    const float* wr = rz_w + o * 64;
    for (int e = 0; e < kD; ++e) acc = fmaf(xin[e], wr[e], fmaf(st[e], wr[32 + e], acc));
    r[o] = sigmoidf_(acc);
  }
  for (int o = 0; o < kD; ++o) {
    float az = rz_b[32 + o], ah = hc_b[o];
    const float* wz = rz_w + (32 + o) * 64;
    const float* wh = hc_w + o * 64;
    for (int e = 0; e < kD; ++e) {
      az = fmaf(xin[e], wz[e], fmaf(st[e], wz[32 + e], az));
      ah = fmaf(xin[e], wh[e], fmaf(r[e] * st[e], wh[32 + e], ah));
    }
    float z = sigmoidf_(az);
    float ns = z * st[o] + (1.0f - z) * tanhf(ah);
    sp[o] = ns;
    h[(size_t)gid * kD + o] = fmaxf(ns, 0.0f);
  }
}

// ---------------------------------------------------------------- kernel 6: fc2 split (a/c for sup & mask)
__global__ void fc2_kernel(const float* __restrict__ h,
                           const float* __restrict__ wS, const float* __restrict__ wM,
                           float* aS, float* cS, float* aM, float* cM)
{
  int gid = blockIdx.x * blockDim.x + threadIdx.x;
  if (gid >= kB * kN * kD) return;
  int d = gid % kD;
  const float* hp = h + (size_t)(gid / kD) * kD;
  float s1 = 0.f, s2 = 0.f, s3 = 0.f, s4 = 0.f;
  for (int e = 0; e < kD; ++e) {
    float hv = hp[e];
    s1 = fmaf(hv, wS[d * 64 + e], s1);
    s2 = fmaf(hv, wS[d * 64 + 32 + e], s2);
    s3 = fmaf(hv, wM[d * 64 + e], s3);
    s4 = fmaf(hv, wM[d * 64 + 32 + e], s4);
  }
  aS[gid] = s1; cS[gid] = s2; aM[gid] = s3; cM[gid] = s4;
}

// ---------------------------------------------------------------- kernel 7: fused ReLU-dot adjacency
// adj[b,v,w] = (sum_d relu(aS+cS+bS)*fS + fbS) * sigmoid(sum_d relu(aM+cM+bM)*fM + fbM)
__global__ void adj_kernel(const float* __restrict__ aS, const float* __restrict__ cS,
                           const float* __restrict__ aM, const float* __restrict__ cM,
                           const float* __restrict__ bS, const float* __restrict__ fS, const float* fbS,
                           const float* __restrict__ bM, const float* __restrict__ fM, const float* fbM,
                           float* __restrict__ adj)
{
  __shared__ float sAS[kD], sAM[kD], sFS[kD], sFM[kD];
  int bv = blockIdx.x;  // b*kN + v
  int tid = threadIdx.x;
  if (tid < kD) {
    sAS[tid] = aS[(size_t)bv * kD + tid] + bS[tid];
    sAM[tid] = aM[(size_t)bv * kD + tid] + bM[tid];
    sFS[tid] = fS[tid];
    sFM[tid] = fM[tid];
  }
  __syncthreads();
  int w = tid;
  if (w < kN) {
    int b = bv / kN;
    const float* cSp = cS + ((size_t)b * kN + w) * kD;
    const float* cMp = cM + ((size_t)b * kN + w) * kD;
    float ss = fbS[0], sm = fbM[0];
    for (int d = 0; d < kD; ++d) {
      ss = fmaf(fmaxf(sAS[d] + cSp[d], 0.0f), sFS[d], ss);
      sm = fmaf(fmaxf(sAM[d] + cMp[d], 0.0f), sFM[d], sm);
    }
    adj[(size_t)bv * kN + w] = ss * sigmoidf_(sm);
  }
}

// ---------------------------------------------------------------- kernel 8: copy x_i chunk into ho[g=0]
__global__ void copy_xi_kernel(const float* __restrict__ xt, float* __restrict__ ho, int chunk)
{
  int gid = blockIdx.x * blockDim.x + threadIdx.x;
  if (gid >= kB * kConv * kN * kIntv) return;
  int t = gid % kIntv;
  int n = (gid / kIntv) % kN;
  int c = (gid / (kIntv * kN)) % kConv;
  int b = gid / (kIntv * kN * kConv);
  ho[(((size_t)b * 96 + c) * kN + n) * kIntv + t] =
      xt[(((size_t)b * kConv + c) * kN + n) * kTlen + chunk * kIntv + t];
}

// ---------------------------------------------------------------- kernel 9: mixprop hop (WMMA f32 GEMM)
// hout[b,c,w,t] = alpha*xc[b,c,w,t] + (1-alpha)*sum_v hin[b,c,v,t]*adj[b,v,w]
// M = c*16+t (512), N = w (200, padded to 208), K = v (200)
__global__ void mixprop_wmma_kernel(float* __restrict__ ho, const float* __restrict__ adj,
                                    int gIn, int gOut)
{
  int lane = threadIdx.x & 31, l16 = lane & 15, half = lane >> 4;
  int mt = blockIdx.x;  // 0..31
  int nt = blockIdx.y;  // 0..12
  int b  = blockIdx.z;
  const float* hin  = ho + ((size_t)b * 96 + gIn * 32) * kN * kIntv;
  const float* xc   = ho + ((size_t)b * 96) * kN * kIntv;          // g0 = x_i
  float*       hout = ho + ((size_t)b * 96 + gOut * 32) * kN * kIntv;
  const float* adjb = adj + (size_t)b * kN * kN;
  int mA = mt * 16 + l16, cA = mA >> 4, tA = mA & 15;
  int col = nt * 16 + l16;
  int colc = (col < kN) ? col : (kN - 1);   // clamp loads; stores masked below
  v8f acc = {};
  for (int v0 = 0; v0 < kN; v0 += 4) {
    int ka = v0 + 2 * half;
    v2f a, bv;
    a.x = hin[(cA * kN + ka) * kIntv + tA];
    a.y = hin[(cA * kN + ka + 1) * kIntv + tA];
    bv.x = adjb[(size_t)ka * kN + colc];
    bv.y = adjb[(size_t)(ka + 1) * kN + colc];
    acc = wmma_f32(a, bv, acc);
  }
  int w = nt * 16 + l16;
  for (int r = 0; r < 8; ++r) {
    int m = mt * 16 + r + 8 * half;
    int c = m >> 4, t = m & 15;
    if (w < kN)
      hout[(c * kN + w) * kIntv + t] =
          kAlpha * xc[(c * kN + w) * kIntv + t] + (1.0f - kAlpha) * acc[r];
  }
}

// ---------------------------------------------------------------- kernel 10: MLP over concat(h0,h1,h2) + residual (WMMA f32)
// out[b,o,w,chunk*16+t] = sum_{k<96} ho[b,k,w,t]*mlp_w[o,k] + mlp_b[o] + x[b,o,w,6+chunk*16+t]
__global__ void mlp_wmma_kernel(const float* __restrict__ ho, const float* __restrict__ mlp_w,
                                const float* __restrict__ mlp_b, const float* __restrict__ x,
                                float* __restrict__ out, int chunk)
{
  int lane = threadIdx.x & 31, l16 = lane & 15, half = lane >> 4;
  int mt = blockIdx.x;  // 0..199 : rows (w*16+t)
  int ct = blockIdx.y;  // 0..1   : o tiles
  int b  = blockIdx.z;
  const float* hob = ho + (size_t)b * 96 * kN * kIntv;
  int mA = mt * 16 + l16, wA = mA >> 4, tA = mA & 15;
  int col = ct * 16 + l16;  // o
  v8f acc = {};
  for (int k0 = 0; k0 < 96; k0 += 4) {
    int ka = k0 + 2 * half;
    v2f a, bv;
    a.x = hob[(ka * kN + wA) * kIntv + tA];
    a.y = hob[((ka + 1) * kN + wA) * kIntv + tA];
    bv.x = mlp_w[col * 96 + ka];
    bv.y = mlp_w[col * 96 + ka + 1];
    acc = wmma_f32(a, bv, acc);
  }
  for (int r = 0; r < 8; ++r) {
    int m = mt * 16 + r + 8 * half;
    int w = m >> 4, t = m & 15;
    int tg = chunk * kIntv + t;
    float val = acc[r] + mlp_b[col] +
                x[(((size_t)b * kRes + col) * kN + w) * kTin + 6 + tg];
    out[(((size_t)b * kRes + col) * kN + w) * kTlen + tg] = val;
  }
}

// ---------------------------------------------------------------- LayerNorm (deterministic two-stage)
__global__ void ln_reduce1_kernel(const float* __restrict__ out, float* partS, float* partQ)
{
  __shared__ float ss[256], sq[256];
  int b = blockIdx.x, s = blockIdx.y;
  const float* p = out + (size_t)b * kPerB + (size_t)s * (kPerB / 64);
  float acc = 0.f, acc2 = 0.f;
  for (int i = threadIdx.x; i < kPerB / 64; i += 256) {
    float v = p[i];
    acc += v; acc2 += v * v;
  }
  ss[threadIdx.x] = acc; sq[threadIdx.x] = acc2;
  __syncthreads();
  for (int off = 128; off > 0; off >>= 1) {
    if (threadIdx.x < off) { ss[threadIdx.x] += ss[threadIdx.x + off]; sq[threadIdx.x] += sq[threadIdx.x + off]; }
    __syncthreads();
  }
  if (threadIdx.x == 0) { partS[b * 64 + s] = ss[0]; partQ[b * 64 + s] = sq[0]; }
}

__global__ void ln_reduce2_kernel(const float* partS, const float* partQ, float* mv)
{
  int b = threadIdx.x;
  if (b >= kB) return;
  float s = 0.f, q = 0.f;
  for (int i = 0; i < 64; ++i) { s += partS[b * 64 + i]; q += partQ[b * 64 + i]; }
  float mean = s / (float)kPerB;
  float var  = q / (float)kPerB - mean * mean;
  mv[2 * b] = mean;
  mv[2 * b + 1] = rsqrtf(var + kEps);
}

__global__ void ln_apply_kernel(float* __restrict__ out, const float* __restrict__ mv,
                                const float* __restrict__ lnw, const float* __restrict__ lnb)
{
  int idx = blockIdx.x * blockDim.x + threadIdx.x;
  if (idx >= kXnElems) return;
  int b = idx / kPerB;
  int rem = idx % kPerB;  // ((c*200)+n)*64 + t  == ln_w flat index
  float v = out[idx];
  out[idx] = (v - mv[2 * b]) * mv[2 * b + 1] * lnw[rem] + lnb[rem];
}

// ---------------------------------------------------------------- launch
extern "C" void kernel_launch(void* const* d_in, const int* in_sizes, int n_in,
                              void* d_out, int out_size, void* d_ws, size_t ws_size,
                              hipStream_t stream)
{
  const float* x    = (const float*)d_in[0];
  const float* stat = (const float*)d_in[1];
  const float* fw[4] = {(const float*)d_in[2], (const float*)d_in[4], (const float*)d_in[6], (const float*)d_in[8]};
  const float* fb[4] = {(const float*)d_in[3], (const float*)d_in[5], (const float*)d_in[7], (const float*)d_in[9]};
  const float* gw[4] = {(const float*)d_in[10], (const float*)d_in[12], (const float*)d_in[14], (const float*)d_in[16]};
  const float* gb[4] = {(const float*)d_in[11], (const float*)d_in[13], (const float*)d_in[15], (const float*)d_in[17]};
  const float* skip_w = (const float*)d_in[18];
  const float* skip_b = (const float*)d_in[19];
  const float* s2d_w  = (const float*)d_in[20];
  const float* s2d_b  = (const float*)d_in[21];
  const float* rz_w   = (const float*)d_in[22];
  const float* rz_b   = (const float*)d_in[23];
  const float* hc_w   = (const float*)d_in[24];
  const float* hc_b   = (const float*)d_in[25];
  const float* supW2  = (const float*)d_in[26];
  const float* supB2  = (const float*)d_in[27];
  const float* supW1  = (const float*)d_in[28];
  const float* supB1  = (const float*)d_in[29];
  const float* mskW2  = (const float*)d_in[30];
  const float* mskB2  = (const float*)d_in[31];
  const float* mskW1  = (const float*)d_in[32];
  const float* mskB1  = (const float*)d_in[33];
  const float* mlp_w  = (const float*)d_in[34];
  const float* mlp_b  = (const float*)d_in[35];
  const float* ln_w   = (const float*)d_in[36];
  const float* ln_b   = (const float*)d_in[37];

  float* ws    = (float*)d_ws;
  float* xt    = ws + OFF_XT;
  float* inp   = ws + OFF_INP;
  float* state = ws + OFF_STATE;
  float* h     = ws + OFF_H;
  float* aS    = ws + OFF_AS;
  float* cS    = ws + OFF_CS;
  float* aM    = ws + OFF_AM;
  float* cM    = ws + OFF_CM;
  float* adj   = ws + OFF_ADJ;
  float* ho    = ws + OFF_HO;
  float* partS = ws + OFF_PS;
  float* partQ = ws + OFF_PQ;
  float* mv    = ws + OFF_MV;

  float* xn   = (float*)d_out;
  float* skip = xn + kXnElems;

  // 1) gated dilated conv -> xt
  conv_gated_kernel<<<(kXnElems + 255) / 256, 256, 0, stream>>>(
      x, fw[0], fw[1], fw[2], fw[3], fb[0], fb[1], fb[2], fb[3],
      gw[0], gw[1], gw[2], gw[3], gb[0], gb[1], gb[2], gb[3], xt);

  // 2) skip conv (WMMA)
  skip_wmma_kernel<<<dim3(kB * kN / 16, kSkip / 16), 32, 0, stream>>>(xt, skip_w, skip_b, skip);

  // 3) chunk means + initial state
  chunk_means_kernel<<<(kB * kNCh * kN * kD + 255) / 256, 256, 0, stream>>>(xt, inp);
  state_init_kernel<<<(kB * kN * kD + 255) / 256, 256, 0, stream>>>(stat, s2d_w, s2d_b, state);

  // 4) sequential chunks
  for (int ch = 0; ch < kNCh; ++ch) {
    gru_kernel<<<(kB * kN + 63) / 64, 64, 0, stream>>>(inp, state, rz_w, rz_b, hc_w, hc_b, h, ch);
    fc2_kernel<<<(kB * kN * kD + 255) / 256, 256, 0, stream>>>(h, supW2, mskW2, aS, cS, aM, cM);
    adj_kernel<<<kB * kN, 224, 0, stream>>>(aS, cS, aM, cM, supB2, supW1, supB1,
                                            mskB2, mskW1, mskB1, adj);
    copy_xi_kernel<<<(kB * kConv * kN * kIntv + 255) / 256, 256, 0, stream>>>(xt, ho, ch);
    mixprop_wmma_kernel<<<dim3(32, 13, kB), 32, 0, stream>>>(ho, adj, 0, 1);
    mixprop_wmma_kernel<<<dim3(32, 13, kB), 32, 0, stream>>>(ho, adj, 1, 2);
    mlp_wmma_kernel<<<dim3(kN, 2, kB), 32, 0, stream>>>(ho, mlp_w, mlp_b, x, xn, ch);
  }

  // 5) LayerNorm over (C,N,T) per batch, deterministic fixed-order reduction
  ln_reduce1_kernel<<<dim3(kB, 64), 256, 0, stream>>>(xn, partS, partQ);
  ln_reduce2_kernel<<<1, 32, 0, stream>>>(partS, partQ, mv);
  ln_apply_kernel<<<(kXnElems + 255) / 256, 256, 0, stream>>>(xn, mv, ln_w, ln_b);
}